// GIN_26963804684653
// MI455X (gfx1250) — compile-verified
//
#include <hip/hip_runtime.h>
#include <cmath>

// ---------------------------------------------------------------------------
// GIN forward for MI455X (gfx1250, wave32).
//  - scatter-add via inline-asm global_atomic_add_f32 scope:SCOPE_DEV
//    (guaranteed native single-instruction fp32 atomic; dest arrays are
//    L2-resident: 25.6MB / 12.8MB << 192MB L2, so atomics run at L2 speed;
//    only the edge list streams from HBM).
//  - all N x {64,32} -> 32 GEMMs use V_WMMA_F32_16X16X4_F32 (fp32 WMMA) so we
//    keep exact fp32 semantics while using the CDNA5 matrix pipe.
// ---------------------------------------------------------------------------

typedef __attribute__((ext_vector_type(2))) float v2f;
typedef __attribute__((ext_vector_type(8))) float v8f;

#define BN_EPS 1e-5f

__device__ __forceinline__ void atomic_add_f32_dev(float* p, float v)
{
    // GLOBAL_ATOMIC_ADD_F32, no-return (STOREcnt), device scope -> RMW at L2.
    asm volatile("global_atomic_add_f32 %0, %1, off scope:SCOPE_DEV"
                 :
                 : "v"(p), "v"(v)
                 : "memory");
}

// Fold BatchNorm (eval) + linear bias into per-column scale/bias:
//   bn(z + b_lin) = z*scale + bias,  scale = g*rsqrt(v+eps),
//   bias = (b_lin - m)*scale + beta
__global__ void bn_prep_kernel(
    const float* g1, const float* be1, const float* m1, const float* v1, const float* lb1,
    const float* g2, const float* be2, const float* m2, const float* v2, const float* lb2,
    const float* g3, const float* be3, const float* m3, const float* v3, const float* lb3,
    float* bn)  // [3][64] = {scale[32], bias[32]} per layer
{
    int t = threadIdx.x;
    if (t >= 96) return;
    int l = t >> 5, j = t & 31;
    const float* g  = (l == 0) ? g1  : (l == 1) ? g2  : g3;
    const float* be = (l == 0) ? be1 : (l == 1) ? be2 : be3;
    const float* m  = (l == 0) ? m1  : (l == 1) ? m2  : m3;
    const float* v  = (l == 0) ? v1  : (l == 1) ? v2  : v3;
    const float* lb = (l == 0) ? lb1 : (l == 1) ? lb2 : lb3;
    float s = g[j] * rsqrtf(v[j] + BN_EPS);
    bn[l * 64 + j]      = s;
    bn[l * 64 + 32 + j] = (lb[j] - m[j]) * s + be[j];
}

__global__ void copy4_kernel(const float4* __restrict__ src,
                             float4* __restrict__ dst, int n4)
{
    int i = blockIdx.x * blockDim.x + threadIdx.x;
    if (i < n4) dst[i] = src[i];
}

// acc[dst[e], :] += feat[src[e], :] — 4 features per thread, float4 gather,
// native fp32 atomics at device scope (performed in L2).
template <int F>
__global__ void scatter_add_kernel(const float* __restrict__ feat,
                                   const int* __restrict__ ei,
                                   float* __restrict__ acc, int E)
{
    constexpr int CH = F / 4;  // float4 chunks per edge
    int tid = blockIdx.x * blockDim.x + threadIdx.x;
    if (tid >= E * CH) return;
    int e = tid / CH;
    int c = (tid - e * CH) * 4;
    int s = ei[e];
    int d = ei[E + e];
    const float4 v = *reinterpret_cast<const float4*>(feat + (size_t)s * F + c);
    float* dp = acc + (size_t)d * F + c;
    atomic_add_f32_dev(dp + 0, v.x);
    atomic_add_f32_dev(dp + 1, v.y);
    atomic_add_f32_dev(dp + 2, v.z);
    atomic_add_f32_dev(dp + 3, v.w);
}

// out[N,32] = relu((in[N,K] @ W[K,32]) * scale + bias); scale==nullptr -> 1.
// One wave computes a 16x32 tile with V_WMMA_F32_16X16X4_F32.
// A frag (16x4 f32):  lane<16: M=lane, {K=kb+0, kb+1}; lane>=16: {K=kb+2, kb+3}
// B frag (4x16 f32):  symmetric over K halves; col = 16*t + (lane&15)
// C/D frag: vgpr r -> row = r + 8*(lane>>4), col = tilebase + (lane&15)
template <int K>
__global__ __launch_bounds__(128) void gemm_bn_relu_kernel(
    const float* __restrict__ in, const float* __restrict__ W,
    const float* __restrict__ scale, const float* __restrict__ bias,
    float* __restrict__ out, int nrows)
{
    __shared__ v2f Wp[(K / 2) * 32];  // Wp[p][n] = (W[2p][n], W[2p+1][n])
    int tid = threadIdx.x;
    for (int idx = tid; idx < (K / 2) * 32; idx += 128) {
        int p = idx >> 5, n = idx & 31;
        v2f w;
        w.x = W[(2 * p) * 32 + n];
        w.y = W[(2 * p + 1) * 32 + n];
        Wp[idx] = w;
    }
    __syncthreads();

    int wave = tid >> 5;
    int lane = tid & 31;
    int tile = blockIdx.x * 4 + wave;
    if (tile * 16 >= nrows) return;

    int m  = lane & 15;
    int hi = lane >> 4;
    int row = tile * 16 + m;
    int rowc = (row < nrows) ? row : (nrows - 1);  // clamp (stores are guarded)
    const float* arow = in + (size_t)rowc * K + 2 * hi;

    v8f c0 = {};
    v8f c1 = {};
#pragma unroll
    for (int kb = 0; kb < K; kb += 4) {
        v2f a;
        a.x = arow[kb];
        a.y = arow[kb + 1];
        int p = (kb >> 1) + hi;
        v2f b0 = Wp[p * 32 + m];        // cols 0..15
        v2f b1 = Wp[p * 32 + 16 + m];   // cols 16..31
        c0 = __builtin_amdgcn_wmma_f32_16x16x4_f32(false, a, false, b0,
                                                   (short)0, c0, false, false);
        c1 = __builtin_amdgcn_wmma_f32_16x16x4_f32(false, a, false, b1,
                                                   (short)0, c1, false, false);
    }

    float sc0 = scale ? scale[m]      : 1.0f;
    float sc1 = scale ? scale[16 + m] : 1.0f;
    float bi0 = bias[m];
    float bi1 = bias[16 + m];
#pragma unroll
    for (int r = 0; r < 8; ++r) {
        int orow = tile * 16 + r + 8 * hi;
        if (orow < nrows) {
            float v0 = fmaxf(c0[r] * sc0 + bi0, 0.0f);
            float v1 = fmaxf(c1[r] * sc1 + bi1, 0.0f);
            out[(size_t)orow * 32 + m]      = v0;
            out[(size_t)orow * 32 + 16 + m] = v1;
        }
    }
}

__global__ void rowsum32_kernel(const float* __restrict__ in,
                                float* __restrict__ s, int n)
{
    int i = blockIdx.x * blockDim.x + threadIdx.x;
    if (i >= n) return;
    const float4* r = reinterpret_cast<const float4*>(in + (size_t)i * 32);
    float acc = 0.0f;
#pragma unroll
    for (int q = 0; q < 8; ++q) {
        float4 v = r[q];
        acc += v.x + v.y + v.z + v.w;
    }
    s[i] = acc;
}

__global__ void lin32to1_relu_kernel(const float* __restrict__ in,
                                     const float* __restrict__ w,
                                     const float* __restrict__ b,
                                     float* __restrict__ out, int n)
{
    int i = blockIdx.x * blockDim.x + threadIdx.x;
    if (i >= n) return;
    const float* r = in + (size_t)i * 32;
    float acc = b[0];
#pragma unroll
    for (int j = 0; j < 32; ++j) acc += r[j] * w[j];
    out[i] = fmaxf(acc, 0.0f);
}

// feat = [s1, s2, s3, sigmoid(s3)]; out = relu(feat@fc1 + b1) @ fc2 + b2
__global__ void head_kernel(const float* __restrict__ s1, const float* __restrict__ s2,
                            const float* __restrict__ s3,
                            const float* __restrict__ fc1w, const float* __restrict__ fc1b,
                            const float* __restrict__ fc2w, const float* __restrict__ fc2b,
                            float* __restrict__ out, int n)
{
    int i = blockIdx.x * blockDim.x + threadIdx.x;
    if (i >= n) return;
    float a = s1[i], b = s2[i], c = s3[i];
    float sg = 1.0f / (1.0f + expf(-c));
    float acc = fc2b[0];
#pragma unroll
    for (int j = 0; j < 32; ++j) {
        float r = a * fc1w[j] + b * fc1w[32 + j] + c * fc1w[64 + j] +
                  sg * fc1w[96 + j] + fc1b[j];
        acc += fmaxf(r, 0.0f) * fc2w[j];
    }
    out[i]     = acc;  // r
    out[n + i] = c;    // s3
}

extern "C" void kernel_launch(void* const* d_in, const int* in_sizes, int n_in,
                              void* d_out, int out_size, void* d_ws, size_t ws_size,
                              hipStream_t stream)
{
    const float* x    = (const float*)d_in[0];
    const int*   ei   = (const int*)d_in[1];
    const float* w1_1 = (const float*)d_in[2];  const float* b1_1 = (const float*)d_in[3];
    const float* bn1g = (const float*)d_in[4];  const float* bn1b = (const float*)d_in[5];
    const float* bn1m = (const float*)d_in[6];  const float* bn1v = (const float*)d_in[7];
    const float* w1_2 = (const float*)d_in[8];  const float* b1_2 = (const float*)d_in[9];
    const float* w2_1 = (const float*)d_in[10]; const float* b2_1 = (const float*)d_in[11];
    const float* bn2g = (const float*)d_in[12]; const float* bn2b = (const float*)d_in[13];
    const float* bn2m = (const float*)d_in[14]; const float* bn2v = (const float*)d_in[15];
    const float* w2_2 = (const float*)d_in[16]; const float* b2_2 = (const float*)d_in[17];
    const float* w3_1 = (const float*)d_in[18]; const float* b3_1 = (const float*)d_in[19];
    const float* bn3g = (const float*)d_in[20]; const float* bn3b = (const float*)d_in[21];
    const float* bn3m = (const float*)d_in[22]; const float* bn3v = (const float*)d_in[23];
    const float* w3_2 = (const float*)d_in[24]; const float* b3_2 = (const float*)d_in[25];
    const float* fc1w = (const float*)d_in[26]; const float* fc1b = (const float*)d_in[27];
    const float* fc2w = (const float*)d_in[28]; const float* fc2b = (const float*)d_in[29];

    const int N = in_sizes[0] / 64;
    const int E = in_sizes[1] / 2;

    // Workspace layout (floats), ~65 MB total:
    float* ws = (float*)d_ws;
    float* A1 = ws;                        // [N,64] conv1 (x + agg)
    float* X1 = ws + (size_t)N * 64;       // [N,32]
    float* X2 = ws + (size_t)N * 96;       // [N,32]
    float* HT = ws + (size_t)N * 128;      // [N,32] MLP hidden temp
    float* S1 = ws + (size_t)N * 160;      // [N]
    float* S2 = S1 + N;                    // [N]
    float* S3 = S2 + N;                    // [N]
    float* BN = S3 + N;                    // [3][64] folded scale/bias
    float* A2 = A1;                        // reuse A1 low half  [N,32]
    float* A3 = A1 + (size_t)N * 32;       // reuse A1 high half [N,32]
    float* out = (float*)d_out;

    const int T = 256;
    const int tiles   = (N + 15) / 16;
    const int gblocks = (tiles + 3) / 4;   // 4 waves (tiles) per block

    bn_prep_kernel<<<1, 96, 0, stream>>>(bn1g, bn1b, bn1m, bn1v, b1_1,
                                         bn2g, bn2b, bn2m, bn2v, b2_1,
                                         bn3g, bn3b, bn3m, bn3v, b3_1, BN);

    // ---- conv1: h = x + segment_sum(x[src], dst); x1 = mlp1(h)
    copy4_kernel<<<(N * 16 + T - 1) / T, T, 0, stream>>>((const float4*)x, (float4*)A1, N * 16);
    scatter_add_kernel<64><<<(E * 16 + T - 1) / T, T, 0, stream>>>(x, ei, A1, E);
    gemm_bn_relu_kernel<64><<<gblocks, 128, 0, stream>>>(A1, w1_1, BN,      BN + 32,  HT, N);
    gemm_bn_relu_kernel<32><<<gblocks, 128, 0, stream>>>(HT, w1_2, nullptr, b1_2,     X1, N);
    rowsum32_kernel<<<(N + T - 1) / T, T, 0, stream>>>(X1, S1, N);

    // ---- conv2
    copy4_kernel<<<(N * 8 + T - 1) / T, T, 0, stream>>>((const float4*)X1, (float4*)A2, N * 8);
    scatter_add_kernel<32><<<(E * 8 + T - 1) / T, T, 0, stream>>>(X1, ei, A2, E);
    gemm_bn_relu_kernel<32><<<gblocks, 128, 0, stream>>>(A2, w2_1, BN + 64, BN + 96,  HT, N);
    gemm_bn_relu_kernel<32><<<gblocks, 128, 0, stream>>>(HT, w2_2, nullptr, b2_2,     X2, N);
    rowsum32_kernel<<<(N + T - 1) / T, T, 0, stream>>>(X2, S2, N);

    // ---- conv3 (final linear is 32 -> 1)
    copy4_kernel<<<(N * 8 + T - 1) / T, T, 0, stream>>>((const float4*)X2, (float4*)A3, N * 8);
    scatter_add_kernel<32><<<(E * 8 + T - 1) / T, T, 0, stream>>>(X2, ei, A3, E);
    gemm_bn_relu_kernel<32><<<gblocks, 128, 0, stream>>>(A3, w3_1, BN + 128, BN + 160, HT, N);
    lin32to1_relu_kernel<<<(N + T - 1) / T, T, 0, stream>>>(HT, w3_2, b3_2, S3, N);

    // ---- head: out = [r ; s3]
    head_kernel<<<(N + T - 1) / T, T, 0, stream>>>(S1, S2, S3, fc1w, fc1b, fc2w, fc2b, out, N);
}